// HGIB_Context_Model_9500467658892
// MI455X (gfx1250) — compile-verified
//
#include <hip/hip_runtime.h>
#include <hip/hip_bf16.h>

typedef __attribute__((ext_vector_type(2))) float v2f;
typedef __attribute__((ext_vector_type(8))) float v8f;

#define Uc 100000
#define Dc 10000
#define Ec 500000
#define Lc 100000
#define Hc 128
#define Oc 64

// ---- small-buffer layout (floats) ----
#define SM_TACC1 0
#define SM_TTR1  32
#define SM_TSE1  64
#define SM_TACC2 96
#define SM_TTR2  128
#define SM_TSE2  160
#define SM_NV1   192
#define SM_NV2   196
#define SM_C1    200
#define SM_C2    201
#define SM_VS1V  256
#define SM_VD1V  384
#define SM_VS1R  512
#define SM_VD1R  640
#define SM_VS2V  768
#define SM_VD2V  896
#define SM_VS2R  1024
#define SM_VD2R  1152
#define SM_TOTAL 2048

__device__ inline float dot128(const float* a, const float* b) {
    float s = 0.f;
    for (int i = 0; i < 128; ++i) s += a[i] * b[i];
    return s;
}
__device__ inline float dot32(const float* a, const float* b) {
    float s = 0.f;
    for (int i = 0; i < 32; ++i) s += a[i] * b[i];
    return s;
}

// monotone float <-> uint key for atomic max over floats
__device__ inline unsigned fkeyEnc(float f) {
    unsigned b = __float_as_uint(f);
    return (b & 0x80000000u) ? ~b : (b | 0x80000000u);
}
__device__ inline float fkeyDec(unsigned k) {
    return (k & 0x80000000u) ? __uint_as_float(k ^ 0x80000000u)
                             : __uint_as_float(~k);
}

// ---------------- node encoders ----------------
__global__ __launch_bounds__(256) void encUserK(const float* __restrict__ x,
                                                const float* __restrict__ emb,
                                                const float* __restrict__ Wf,
                                                const float* __restrict__ bf,
                                                float* __restrict__ h) {
    size_t idx = (size_t)blockIdx.x * blockDim.x + threadIdx.x;
    if (idx >= (size_t)Uc * Hc) return;
    size_t n = idx >> 7;
    int hc = (int)(idx & 127);
    h[idx] = emb[idx] + x[n * 3] * Wf[hc] + x[n * 3 + 1] * Wf[128 + hc] +
             x[n * 3 + 2] * Wf[256 + hc] + bf[hc];
}

__global__ __launch_bounds__(256) void encDestK(const float* __restrict__ x,
                                                const float* __restrict__ emb,
                                                const float* __restrict__ Wf,
                                                const float* __restrict__ bf,
                                                float* __restrict__ h) {
    size_t idx = (size_t)blockIdx.x * blockDim.x + threadIdx.x;
    if (idx >= (size_t)Dc * Hc) return;
    size_t n = idx >> 7;
    int hc = (int)(idx & 127);
    h[idx] = emb[idx] + x[n] * Wf[hc] + bf[hc];
}

// ---------------- tiny precompute: projected score vectors + edge lookup tables ----------------
__global__ __launch_bounds__(128) void prepK(
    const float* c1v_We, const float* c1v_ae, const float* c2v_We, const float* c2v_ae,
    const float* Weout, const float* beout,
    const float* emb_acc, const float* emb_trans, const float* emb_season,
    const float* Wnum, const float* bnum,
    const float* c1v_Ws, const float* c1v_as, const float* c1v_Wd, const float* c1v_ad,
    const float* c1r_Ws, const float* c1r_as, const float* c1r_Wd, const float* c1r_ad,
    const float* c2v_W, const float* c2v_as, const float* c2v_ad,
    const float* c2r_W, const float* c2r_as, const float* c2r_ad,
    float* sm) {
    __shared__ float ve[128], w1s[128], w2s[128];
    __shared__ float ceA, ceB;
    int t = threadIdx.x;

    // layer1 edge projection:  w1 = Weout @ (c1v_We @ c1v_ae)
    ve[t] = dot128(c1v_We + t * 128, c1v_ae);
    __syncthreads();
    w1s[t] = dot128(Weout + t * 128, ve);
    if (t == 0) ceA = dot128(beout, ve);
    __syncthreads();
    // layer2 edge projection
    ve[t] = dot128(c2v_We + t * 128, c2v_ae);
    __syncthreads();
    w2s[t] = dot128(Weout + t * 128, ve);
    if (t == 0) ceB = dot128(beout, ve);
    __syncthreads();

    // rank-1 attention score projections:  (h@W)·a == h·(W@a)
    sm[SM_VS1V + t] = dot128(c1v_Ws + t * 128, c1v_as);
    sm[SM_VD1V + t] = dot128(c1v_Wd + t * 128, c1v_ad);
    sm[SM_VS1R + t] = dot128(c1r_Ws + t * 128, c1r_as);
    sm[SM_VD1R + t] = dot128(c1r_Wd + t * 128, c1r_ad);
    sm[SM_VS2V + t] = dot128(c2v_W + t * 128, c2v_as);
    sm[SM_VD2V + t] = dot128(c2v_W + t * 128, c2v_ad);
    sm[SM_VS2R + t] = dot128(c2r_W + t * 128, c2r_as);
    sm[SM_VD2R + t] = dot128(c2r_W + t * 128, c2r_ad);

    // categorical lookup tables (ef never materialized)
    if (t < 20) {
        sm[SM_TACC1 + t] = dot32(emb_acc + t * 32, w1s);
        sm[SM_TTR1 + t]  = dot32(emb_trans + t * 32, w1s + 32);
        sm[SM_TSE1 + t]  = dot32(emb_season + t * 32, w1s + 64);
        sm[SM_TACC2 + t] = dot32(emb_acc + t * 32, w2s);
        sm[SM_TTR2 + t]  = dot32(emb_trans + t * 32, w2s + 32);
        sm[SM_TSE2 + t]  = dot32(emb_season + t * 32, w2s + 64);
    }
    if (t < 3) {
        sm[SM_NV1 + t] = dot32(Wnum + t * 32, w1s + 96);
        sm[SM_NV2 + t] = dot32(Wnum + t * 32, w2s + 96);
    }
    if (t == 0) {
        sm[SM_C1] = ceA + dot32(bnum, w1s + 96);
        sm[SM_C2] = ceB + dot32(bnum, w2s + 96);
    }
}

// ---------------- per-edge alpha scalars from tables ----------------
__global__ __launch_bounds__(256) void edgeScalarK(const int* __restrict__ cat,
                                                   const float* __restrict__ num,
                                                   const float* __restrict__ sm,
                                                   float* __restrict__ e1,
                                                   float* __restrict__ e2) {
    int i = blockIdx.x * blockDim.x + threadIdx.x;
    if (i >= Ec) return;
    int c0 = cat[3 * i], c1 = cat[3 * i + 1], c2 = cat[3 * i + 2];
    float n0 = num[3 * i], n1 = num[3 * i + 1], n2 = num[3 * i + 2];
    e1[i] = sm[SM_TACC1 + c0] + sm[SM_TTR1 + c1] + sm[SM_TSE1 + c2] +
            n0 * sm[SM_NV1] + n1 * sm[SM_NV1 + 1] + n2 * sm[SM_NV1 + 2] + sm[SM_C1];
    e2[i] = sm[SM_TACC2 + c0] + sm[SM_TTR2 + c1] + sm[SM_TSE2 + c2] +
            n0 * sm[SM_NV2] + n1 * sm[SM_NV2 + 1] + n2 * sm[SM_NV2 + 2] + sm[SM_C2];
}

// ---------------- generic utilities ----------------
__global__ __launch_bounds__(256) void zeroK(float* __restrict__ p, size_t n) {
    size_t i = (size_t)blockIdx.x * blockDim.x + threadIdx.x;
    if (i < n) p[i] = 0.f;
}

__global__ __launch_bounds__(256) void rowdotK(const float* __restrict__ X,
                                               const float* __restrict__ v,
                                               float* __restrict__ out, int N) {
    int n = blockIdx.x * blockDim.x + threadIdx.x;
    if (n >= N) return;
    const float4* x4 = (const float4*)(X + (size_t)n * 128);
    const float4* v4 = (const float4*)v;
    float s = 0.f;
    for (int i = 0; i < 32; ++i) {
        float4 a = x4[i], b = v4[i];
        s += a.x * b.x + a.y * b.y + a.z * b.z + a.w * b.w;
    }
    out[n] = s;
}

__global__ __launch_bounds__(256) void biasReluK(const float* __restrict__ agg,
                                                 const float* __restrict__ b,
                                                 float* __restrict__ out, size_t total) {
    size_t idx = (size_t)blockIdx.x * blockDim.x + threadIdx.x;
    if (idx >= total) return;
    float v = agg[idx] + b[idx & 127];
    out[idx] = v > 0.f ? v : 0.f;
}

// ---------------- GAT segment softmax ----------------
__global__ __launch_bounds__(256) void alphaK(const int* __restrict__ src,
                                              const int* __restrict__ dst,
                                              const float* __restrict__ sS,
                                              const float* __restrict__ sD,
                                              const float* __restrict__ eAdd,
                                              float* __restrict__ alpha,
                                              unsigned* __restrict__ amax) {
    int i = blockIdx.x * blockDim.x + threadIdx.x;
    if (i >= Ec) return;
    float a = sS[src[i]] + sD[dst[i]];
    if (eAdd) a += eAdd[i];
    a = a > 0.f ? a : 0.2f * a;  // leaky_relu(0.2)
    alpha[i] = a;
    atomicMax(&amax[dst[i]], fkeyEnc(a));
}

__global__ __launch_bounds__(256) void exK(const int* __restrict__ dst,
                                           float* __restrict__ alpha,
                                           const unsigned* __restrict__ amax,
                                           float* __restrict__ denom) {
    int i = blockIdx.x * blockDim.x + threadIdx.x;
    if (i >= Ec) return;
    int d = dst[i];
    float v = __expf(alpha[i] - fkeyDec(amax[d]));
    alpha[i] = v;  // reuse alpha buffer as ex
    atomicAdd(&denom[d], v);
}

__global__ __launch_bounds__(256) void scatterK(const int* __restrict__ src,
                                                const int* __restrict__ dst,
                                                const float* __restrict__ ex,
                                                const float* __restrict__ denom,
                                                const float* __restrict__ xs,
                                                float* __restrict__ out) {
    size_t total = (size_t)Ec * 32;  // 32 float4 chunks per edge
    for (size_t t = (size_t)blockIdx.x * blockDim.x + threadIdx.x; t < total;
         t += (size_t)gridDim.x * blockDim.x) {
        size_t e = t >> 5;
        int c = (int)(t & 31);
        int s = src[e], d = dst[e];
        float w = ex[e] / (denom[d] + 1e-16f);
        float4 x = ((const float4*)xs)[(size_t)s * 32 + c];
        float* o = out + (size_t)d * 128 + c * 4;
        atomicAdd(o + 0, w * x.x);
        atomicAdd(o + 1, w * x.y);
        atomicAdd(o + 2, w * x.z);
        atomicAdd(o + 3, w * x.w);
    }
}

// ---------------- WMMA f32 GEMM: C[N,NCOL] = A[N,128] @ B[128,NCOL] (+bias) ----------------
// One wave = one 16x16 C tile; K looped in steps of 4 via V_WMMA_F32_16X16X4_F32.
// B pair-packed into LDS: Bs[kk*NCOL+col] = (B[2kk][col], B[2kk+1][col]).
template <int NCOL>
__global__ __launch_bounds__(32 * (NCOL / 16)) void gemmK(const float* __restrict__ A,
                                                          const float* __restrict__ B,
                                                          const float* __restrict__ bias,
                                                          float* __restrict__ C, int nRows) {
    __shared__ v2f Bs[64 * NCOL];
    const int nThreads = 32 * (NCOL / 16);
    int tid = threadIdx.x;
    for (int p = tid; p < 64 * NCOL; p += nThreads) {
        int kk = p / NCOL, col = p % NCOL;
        v2f t;
        t.x = B[(2 * kk) * NCOL + col];
        t.y = B[(2 * kk + 1) * NCOL + col];
        Bs[p] = t;
    }
    __syncthreads();

    int wave = tid >> 5;
    int lane = tid & 31;
    int half = lane >> 4;   // K-half select (ISA A/B VGPR layout)
    int mn = lane & 15;     // M for A, N for B/C
    int col0 = wave * 16;
    int row0 = blockIdx.x * 16;
    if (row0 >= nRows) return;

    const v2f* A2 = (const v2f*)A;
    size_t aBase = (size_t)(row0 + mn) * 64;

    v8f c = {};
    for (int it = 0; it < 32; ++it) {  // K = 128 in steps of 4
        v2f a = A2[aBase + 2 * it + half];              // A[m][4it+2h .. +1]
        v2f b = Bs[(2 * it + half) * NCOL + col0 + mn]; // B[4it+2h .. +1][n]
        c = __builtin_amdgcn_wmma_f32_16x16x4_f32(false, a, false, b, (short)0, c,
                                                  false, false);
    }

    float bc = bias ? bias[col0 + mn] : 0.f;
    for (int j = 0; j < 8; ++j) {
        int r = row0 + j + 8 * half;  // C VGPR j: lanes0-15 M=j, lanes16-31 M=j+8
        C[(size_t)r * NCOL + col0 + mn] = c[j] + bc;
    }
}

// ---------------- decode ----------------
__global__ __launch_bounds__(256) void decodeK(const int* __restrict__ ls,
                                               const int* __restrict__ ld,
                                               const float* __restrict__ muU,
                                               const float* __restrict__ muD,
                                               float* __restrict__ pred) {
    int i = blockIdx.x * blockDim.x + threadIdx.x;
    if (i >= Lc) return;
    const float4* a = (const float4*)(muU + (size_t)ls[i] * 64);
    const float4* b = (const float4*)(muD + (size_t)ld[i] * 64);
    float s = 0.f;
    for (int j = 0; j < 16; ++j) {
        float4 x = a[j], y = b[j];
        s += x.x * y.x + x.y * y.y + x.z * y.z + x.w * y.w;
    }
    pred[i] = s;
}

extern "C" void kernel_launch(void* const* d_in, const int* in_sizes, int n_in,
                              void* d_out, int out_size, void* d_ws, size_t ws_size,
                              hipStream_t stream) {
    const float* x_user   = (const float*)d_in[0];
    const float* x_dest   = (const float*)d_in[1];
    const int* edge_src   = (const int*)d_in[2];
    const int* edge_dst   = (const int*)d_in[3];
    const int* eattr_cat  = (const int*)d_in[4];
    const float* eattr_num= (const float*)d_in[5];
    const int* label_src  = (const int*)d_in[6];
    const int* label_dst  = (const int*)d_in[7];
    const float* user_emb = (const float*)d_in[8];
    const float* dest_emb = (const float*)d_in[9];
    const float* Wuf = (const float*)d_in[10];
    const float* buf = (const float*)d_in[11];
    const float* Wdf = (const float*)d_in[12];
    const float* bdf = (const float*)d_in[13];
    const float* emb_acc    = (const float*)d_in[14];
    const float* emb_trans  = (const float*)d_in[15];
    const float* emb_season = (const float*)d_in[16];
    const float* Wnum  = (const float*)d_in[17];
    const float* bnum  = (const float*)d_in[18];
    const float* Weout = (const float*)d_in[19];
    const float* beout = (const float*)d_in[20];
    const float* c1v_Ws = (const float*)d_in[21];
    const float* c1v_Wd = (const float*)d_in[22];
    const float* c1v_as = (const float*)d_in[23];
    const float* c1v_ad = (const float*)d_in[24];
    const float* c1v_We = (const float*)d_in[25];
    const float* c1v_ae = (const float*)d_in[26];
    const float* c1v_b  = (const float*)d_in[27];
    const float* c1r_Ws = (const float*)d_in[28];
    const float* c1r_Wd = (const float*)d_in[29];
    const float* c1r_as = (const float*)d_in[30];
    const float* c1r_ad = (const float*)d_in[31];
    const float* c1r_b  = (const float*)d_in[32];
    const float* c2v_W  = (const float*)d_in[33];
    const float* c2v_as = (const float*)d_in[34];
    const float* c2v_ad = (const float*)d_in[35];
    const float* c2v_We = (const float*)d_in[36];
    const float* c2v_ae = (const float*)d_in[37];
    const float* c2v_b  = (const float*)d_in[38];
    const float* c2r_W  = (const float*)d_in[39];
    const float* c2r_as = (const float*)d_in[40];
    const float* c2r_ad = (const float*)d_in[41];
    const float* c2r_b  = (const float*)d_in[42];
    const float* Wmu = (const float*)d_in[43];
    const float* bmu = (const float*)d_in[44];
    const float* Wls = (const float*)d_in[45];
    const float* bls = (const float*)d_in[46];

    // ---- workspace bump allocator (all counts are multiples of 4 -> 16B aligned) ----
    float* W = (float*)d_ws;
    size_t off = 0;
    auto alloc = [&](size_t n) { float* p = W + off; off += n; return p; };
    float* h_user = alloc((size_t)Uc * Hc);
    float* h_dest = alloc((size_t)Dc * Hc);
    float* xs     = alloc((size_t)Uc * Hc);
    float* aggU   = alloc((size_t)Uc * Hc);
    float* aggD   = alloc((size_t)Dc * Hc);
    float* scoreS = alloc(Uc);
    float* scoreD = alloc(Uc);
    float* e1b    = alloc(Ec);
    float* e2b    = alloc(Ec);
    float* alpha  = alloc(Ec);
    unsigned* amax = (unsigned*)alloc(Uc);
    float* denom  = alloc(Uc);
    float* small  = alloc(SM_TOTAL);
    (void)ws_size; (void)n_in; (void)in_sizes; (void)out_size;

    // ---- node encoders ----
    encUserK<<<((size_t)Uc * Hc + 255) / 256, 256, 0, stream>>>(x_user, user_emb, Wuf, buf, h_user);
    encDestK<<<((size_t)Dc * Hc + 255) / 256, 256, 0, stream>>>(x_dest, dest_emb, Wdf, bdf, h_dest);

    // ---- precompute projections & edge tables ----
    prepK<<<1, 128, 0, stream>>>(c1v_We, c1v_ae, c2v_We, c2v_ae, Weout, beout,
                                 emb_acc, emb_trans, emb_season, Wnum, bnum,
                                 c1v_Ws, c1v_as, c1v_Wd, c1v_ad,
                                 c1r_Ws, c1r_as, c1r_Wd, c1r_ad,
                                 c2v_W, c2v_as, c2v_ad, c2r_W, c2r_as, c2r_ad, small);
    edgeScalarK<<<(Ec + 255) / 256, 256, 0, stream>>>(eattr_cat, eattr_num, small, e1b, e2b);

    // one GAT direction: softmax-weighted scatter of (hS @ Wsrc) into agg[dst]
    auto gatDir = [&](const float* hS, const float* hD, int nS, int nD,
                      const int* sIdx, const int* dIdx, const float* Wsrc,
                      const float* vS, const float* vD, const float* eAdd, float* agg) {
        zeroK<<<(nD + 255) / 256, 256, 0, stream>>>((float*)amax, nD);
        zeroK<<<(nD + 255) / 256, 256, 0, stream>>>(denom, nD);
        zeroK<<<((size_t)nD * Hc + 255) / 256, 256, 0, stream>>>(agg, (size_t)nD * Hc);
        rowdotK<<<(nS + 255) / 256, 256, 0, stream>>>(hS, vS, scoreS, nS);
        rowdotK<<<(nD + 255) / 256, 256, 0, stream>>>(hD, vD, scoreD, nD);
        alphaK<<<(Ec + 255) / 256, 256, 0, stream>>>(sIdx, dIdx, scoreS, scoreD, eAdd, alpha, amax);
        exK<<<(Ec + 255) / 256, 256, 0, stream>>>(dIdx, alpha, amax, denom);
        gemmK<128><<<nS / 16, 256, 0, stream>>>(hS, Wsrc, nullptr, xs, nS);
        scatterK<<<62500, 256, 0, stream>>>(sIdx, dIdx, alpha, denom, xs, agg);
    };

    // ---- layer 1 ----
    gatDir(h_user, h_dest, Uc, Dc, edge_src, edge_dst, c1v_Ws,
           small + SM_VS1V, small + SM_VD1V, e1b, aggD);
    gatDir(h_dest, h_user, Dc, Uc, edge_dst, edge_src, c1r_Ws,
           small + SM_VS1R, small + SM_VD1R, nullptr, aggU);
    biasReluK<<<((size_t)Uc * Hc + 255) / 256, 256, 0, stream>>>(aggU, c1r_b, h_user, (size_t)Uc * Hc);
    biasReluK<<<((size_t)Dc * Hc + 255) / 256, 256, 0, stream>>>(aggD, c1v_b, h_dest, (size_t)Dc * Hc);

    // ---- layer 2 (shared src/dst weight per direction) ----
    gatDir(h_user, h_dest, Uc, Dc, edge_src, edge_dst, c2v_W,
           small + SM_VS2V, small + SM_VD2V, e2b, aggD);
    gatDir(h_dest, h_user, Dc, Uc, edge_dst, edge_src, c2r_W,
           small + SM_VS2R, small + SM_VD2R, nullptr, aggU);
    biasReluK<<<((size_t)Uc * Hc + 255) / 256, 256, 0, stream>>>(aggU, c2r_b, h_user, (size_t)Uc * Hc);
    biasReluK<<<((size_t)Dc * Hc + 255) / 256, 256, 0, stream>>>(aggD, c2v_b, h_dest, (size_t)Dc * Hc);

    // ---- output heads straight into d_out (pred, mu_u, mu_d, ls_u, ls_d) ----
    float* out_pred = (float*)d_out;
    float* out_muU = out_pred + Lc;
    float* out_muD = out_muU + (size_t)Uc * Oc;
    float* out_lsU = out_muD + (size_t)Dc * Oc;
    float* out_lsD = out_lsU + (size_t)Uc * Oc;

    gemmK<64><<<Uc / 16, 128, 0, stream>>>(h_user, Wmu, bmu, out_muU, Uc);
    gemmK<64><<<Dc / 16, 128, 0, stream>>>(h_dest, Wmu, bmu, out_muD, Dc);
    gemmK<64><<<Uc / 16, 128, 0, stream>>>(h_user, Wls, bls, out_lsU, Uc);
    gemmK<64><<<Dc / 16, 128, 0, stream>>>(h_dest, Wls, bls, out_lsD, Dc);

    decodeK<<<(Lc + 255) / 256, 256, 0, stream>>>(label_src, label_dst, out_muU, out_muD, out_pred);
}